// GAT_22728966930471
// MI455X (gfx1250) — compile-verified
//
#include <hip/hip_runtime.h>
#include <hip/hip_bf16.h>
#include <math.h>

#define N_NODES     20000
#define N_EDGES_RAW 320000
#define N_EDGES     340000   // + self loops
#define N_GRAPHS    64
#define NEG_SLOPE   0.2f

typedef float v2f __attribute__((ext_vector_type(2)));
typedef float v8f __attribute__((ext_vector_type(8)));

// ---------------- helpers ----------------

__device__ __forceinline__ float leaky(float x) { return x > 0.f ? x : NEG_SLOPE * x; }
__device__ __forceinline__ float elu_f(float x) { return x > 0.f ? x : expm1f(x); }

// float atomic max via sign-split integer atomics (works from -inf init)
__device__ __forceinline__ void atomicMaxF32(float* addr, float val) {
    if (val >= 0.0f) atomicMax((int*)addr, __float_as_int(val));
    else             atomicMin((unsigned int*)addr, __float_as_uint(val));
}

__device__ __forceinline__ void edge_sd(const int* __restrict__ ei, int e, int& s, int& d) {
    if (e < N_EDGES_RAW) { s = ei[e]; d = ei[N_EDGES_RAW + e]; }
    else                 { s = d = e - N_EDGES_RAW; }
}

// ---------------- generic fill ----------------

__global__ void fill_f32(float* __restrict__ p, int n, float v) {
    int i = blockIdx.x * blockDim.x + threadIdx.x;
    if (i < n) p[i] = v;
}

// ---------------- layer 1: node transform + attention dots ----------------
// h1[n,h,c] = sum_i x[n,i] * W1[i, h*8+c];  al_s1[n,h] = <h1[n,h,:], a_src1[h,:]>
__global__ void node1_kernel(const float* __restrict__ x, const float* __restrict__ W1,
                             const float* __restrict__ a_src1, const float* __restrict__ a_dst1,
                             float* __restrict__ h1, float* __restrict__ al_s, float* __restrict__ al_d) {
    int t = blockIdx.x * blockDim.x + threadIdx.x;
    if (t >= N_NODES * 8) return;
    int n = t >> 3, h = t & 7;
    float x0 = x[n * 3 + 0], x1 = x[n * 3 + 1], x2 = x[n * 3 + 2];
    float s = 0.f, d = 0.f;
#pragma unroll
    for (int c = 0; c < 8; ++c) {
        int col = h * 8 + c;
        float v = fmaf(x0, W1[col], fmaf(x1, W1[64 + col], x2 * W1[128 + col]));
        h1[n * 64 + col] = v;
        s = fmaf(v, a_src1[col], s);
        d = fmaf(v, a_dst1[col], d);
    }
    al_s[t] = s; al_d[t] = d;
}

// ---------------- layer 1: edge passes (per edge-head) ----------------

__global__ void edge1_pass1(const int* __restrict__ ei, const float* __restrict__ al_s,
                            const float* __restrict__ al_d, float* __restrict__ e1,
                            float* __restrict__ m1) {
    int t = blockIdx.x * blockDim.x + threadIdx.x;
    if (t >= N_EDGES * 8) return;
    int e = t >> 3, h = t & 7, s, d;
    edge_sd(ei, e, s, d);
    float v = leaky(al_s[s * 8 + h] + al_d[d * 8 + h]);
    e1[t] = v;
    atomicMaxF32(&m1[d * 8 + h], v);
}

__global__ void edge1_pass2(const int* __restrict__ ei, const float* __restrict__ m1,
                            float* __restrict__ e1, float* __restrict__ denom1) {
    int t = blockIdx.x * blockDim.x + threadIdx.x;
    if (t >= N_EDGES * 8) return;
    int e = t >> 3, h = t & 7, s, d;
    edge_sd(ei, e, s, d);
    float ex = expf(e1[t] - m1[d * 8 + h]);
    e1[t] = ex;
    atomicAdd(&denom1[d * 8 + h], ex);
}

__global__ void edge1_scatter(const int* __restrict__ ei, const float* __restrict__ e1,
                              const float* __restrict__ denom1, const float* __restrict__ h1,
                              float* __restrict__ out1) {
    int t = blockIdx.x * blockDim.x + threadIdx.x;
    if (t >= N_EDGES * 8) return;
    int e = t >> 3, h = t & 7, s, d;
    edge_sd(ei, e, s, d);
    float alpha = e1[t] / denom1[d * 8 + h];
    const float* hs = h1 + s * 64 + h * 8;
    float* od = out1 + d * 64 + h * 8;
#pragma unroll
    for (int c = 0; c < 8; ++c) atomicAdd(&od[c], alpha * hs[c]);
}

// o1 = elu(out1 + b1) in place
__global__ void act1_kernel(float* __restrict__ o1, const float* __restrict__ b1) {
    int i = blockIdx.x * blockDim.x + threadIdx.x;
    if (i < N_NODES * 64) o1[i] = elu_f(o1[i] + b1[i & 63]);
}

// ---------------- WMMA f32 GEMM: h2[N,512] = o1[N,64] @ W2[64,512] ----------------
// one wave per 16x16 tile; V_WMMA_F32_16X16X4_F32, K=64 -> 16 wmma issues per tile
__global__ void gemm1_wmma(const float* __restrict__ A, const float* __restrict__ B,
                           float* __restrict__ C) {
    int wave = threadIdx.x >> 5, lane = threadIdx.x & 31;
    int mBase = blockIdx.x * 16;
    int nBase = (blockIdx.y * 8 + wave) * 16;
    int l15 = lane & 15, kh = lane >> 4;
    v8f acc = {};
#pragma unroll
    for (int k0 = 0; k0 < 64; k0 += 4) {
        int ka = k0 + 2 * kh;
        const float* arow = A + (size_t)(mBase + l15) * 64 + ka;
        v2f a, b;
        a.x = arow[0];
        a.y = arow[1];
        b.x = B[(size_t)ka * 512 + nBase + l15];
        b.y = B[(size_t)(ka + 1) * 512 + nBase + l15];
        acc = __builtin_amdgcn_wmma_f32_16x16x4_f32(false, a, false, b, (short)0, acc,
                                                    false, false);
    }
#pragma unroll
    for (int r = 0; r < 8; ++r) {
        int row = mBase + r + 8 * kh;
        C[(size_t)row * 512 + nBase + l15] = acc[r];
    }
}

// al_s2[n] = <h2[n,:], a_src2>, al_d2 likewise; one wave32 per node
__global__ void node2_kernel(const float* __restrict__ h2, const float* __restrict__ a_src2,
                             const float* __restrict__ a_dst2, float* __restrict__ al_s,
                             float* __restrict__ al_d) {
    int wave = threadIdx.x >> 5, lane = threadIdx.x & 31;
    int n = blockIdx.x * 8 + wave;
    if (n >= N_NODES) return;
    float s = 0.f, d = 0.f;
    const float* hrow = h2 + (size_t)n * 512;
#pragma unroll 4
    for (int c = lane; c < 512; c += 32) {
        float v = hrow[c];
        s = fmaf(v, a_src2[c], s);
        d = fmaf(v, a_dst2[c], d);
    }
#pragma unroll
    for (int m = 16; m > 0; m >>= 1) {
        s += __shfl_xor(s, m, 32);
        d += __shfl_xor(d, m, 32);
    }
    if (lane == 0) { al_s[n] = s; al_d[n] = d; }
}

// ---------------- layer 2: edge passes (per edge, H=1) ----------------

__global__ void edge2_pass1(const int* __restrict__ ei, const float* __restrict__ al_s,
                            const float* __restrict__ al_d, float* __restrict__ e2,
                            float* __restrict__ m2) {
    int e = blockIdx.x * blockDim.x + threadIdx.x;
    if (e >= N_EDGES) return;
    int s, d;
    edge_sd(ei, e, s, d);
    float v = leaky(al_s[s] + al_d[d]);
    e2[e] = v;
    atomicMaxF32(&m2[d], v);
}

__global__ void edge2_pass2(const int* __restrict__ ei, const float* __restrict__ m2,
                            float* __restrict__ e2, float* __restrict__ denom2) {
    int e = blockIdx.x * blockDim.x + threadIdx.x;
    if (e >= N_EDGES) return;
    int s, d;
    edge_sd(ei, e, s, d);
    float ex = expf(e2[e] - m2[d]);
    e2[e] = ex;
    atomicAdd(&denom2[d], ex);
}

// dominant kernel: one 128-thread block per edge; 512-ch gather + atomic scatter.
// h2/out2 are 41MB each -> L2 resident on MI455X (192MB L2).
__global__ void edge2_scatter(const int* __restrict__ ei, const float* __restrict__ ex2,
                              const float* __restrict__ denom2, const float* __restrict__ h2,
                              float* __restrict__ out2) {
    int e = blockIdx.x;
    int s, d;
    edge_sd(ei, e, s, d);
    float alpha = ex2[e] / denom2[d];
    const float* hs = h2 + (size_t)s * 512;
    float* od = out2 + (size_t)d * 512;
    __builtin_prefetch(hs + threadIdx.x, 0, 0);   // global_prefetch_b8
#pragma unroll
    for (int c = threadIdx.x; c < 512; c += 128)
        atomicAdd(&od[c], alpha * hs[c]);
}

// o2 = elu(out2 + b2); pool into per-graph sums
__global__ void act2_pool(const float* __restrict__ out2, const float* __restrict__ b2,
                          const int* __restrict__ batch, float* __restrict__ sums) {
    int t = blockIdx.x * blockDim.x + threadIdx.x;
    if (t >= N_NODES * 512) return;
    int n = t >> 9, c = t & 511;
    float v = elu_f(out2[t] + b2[c]);
    atomicAdd(&sums[(size_t)batch[n] * 512 + c], v);
}

__global__ void count_kernel(const int* __restrict__ batch, float* __restrict__ cnt) {
    int n = blockIdx.x * blockDim.x + threadIdx.x;
    if (n < N_NODES) atomicAdd(&cnt[batch[n]], 1.0f);
}

__global__ void pooled_kernel(const float* __restrict__ sums, const float* __restrict__ cnt,
                              float* __restrict__ pooled) {
    int t = blockIdx.x * blockDim.x + threadIdx.x;
    if (t >= N_GRAPHS * 512) return;
    pooled[t] = sums[t] / fmaxf(cnt[t >> 9], 1.0f);
}

// ---------------- WMMA f32 GEMM: out[64,512] = pooled[64,512] @ Wo[512,512] + bo ----
__global__ void gemm2_wmma(const float* __restrict__ A, const float* __restrict__ B,
                           const float* __restrict__ bias, float* __restrict__ C) {
    int wave = threadIdx.x >> 5, lane = threadIdx.x & 31;
    int t = blockIdx.x * 8 + wave;              // 128 tiles: 4 (M) x 32 (N)
    int mBase = (t >> 5) * 16, nBase = (t & 31) * 16;
    int l15 = lane & 15, kh = lane >> 4;
    v8f acc = {};
#pragma unroll 8
    for (int k0 = 0; k0 < 512; k0 += 4) {
        int ka = k0 + 2 * kh;
        const float* arow = A + (size_t)(mBase + l15) * 512 + ka;
        v2f a, b;
        a.x = arow[0];
        a.y = arow[1];
        b.x = B[(size_t)ka * 512 + nBase + l15];
        b.y = B[(size_t)(ka + 1) * 512 + nBase + l15];
        acc = __builtin_amdgcn_wmma_f32_16x16x4_f32(false, a, false, b, (short)0, acc,
                                                    false, false);
    }
#pragma unroll
    for (int r = 0; r < 8; ++r) {
        int row = mBase + r + 8 * kh;
        C[(size_t)row * 512 + nBase + l15] = acc[r] + bias[nBase + l15];
    }
}

// ---------------- launch ----------------

extern "C" void kernel_launch(void* const* d_in, const int* in_sizes, int n_in,
                              void* d_out, int out_size, void* d_ws, size_t ws_size,
                              hipStream_t stream) {
    (void)in_sizes; (void)n_in; (void)out_size; (void)ws_size;
    const float* x      = (const float*)d_in[0];
    const int*   ei     = (const int*)  d_in[1];
    const int*   batch  = (const int*)  d_in[2];
    const float* W1     = (const float*)d_in[3];
    const float* a_src1 = (const float*)d_in[4];
    const float* a_dst1 = (const float*)d_in[5];
    const float* b1     = (const float*)d_in[6];
    const float* W2     = (const float*)d_in[7];
    const float* a_src2 = (const float*)d_in[8];
    const float* a_dst2 = (const float*)d_in[9];
    const float* b2     = (const float*)d_in[10];
    const float* Wo     = (const float*)d_in[11];
    const float* bo     = (const float*)d_in[12];
    float* out = (float*)d_out;

    float* ws = (float*)d_ws;
    size_t off = 0;
    auto alloc = [&](size_t n) { float* p = ws + off; off += n; return p; };
    float* h1     = alloc((size_t)N_NODES * 64);
    float* al_s1  = alloc((size_t)N_NODES * 8);
    float* al_d1  = alloc((size_t)N_NODES * 8);
    float* e1     = alloc((size_t)N_EDGES * 8);
    float* m1     = alloc((size_t)N_NODES * 8);
    float* denom1 = alloc((size_t)N_NODES * 8);
    float* out1   = alloc((size_t)N_NODES * 64);   // becomes o1 in place
    float* h2     = alloc((size_t)N_NODES * 512);
    float* al_s2  = alloc((size_t)N_NODES);
    float* al_d2  = alloc((size_t)N_NODES);
    float* e2     = alloc((size_t)N_EDGES);
    float* m2     = alloc((size_t)N_NODES);
    float* denom2 = alloc((size_t)N_NODES);
    float* out2   = alloc((size_t)N_NODES * 512);
    float* sums   = alloc((size_t)N_GRAPHS * 512);
    float* cnt    = alloc((size_t)N_GRAPHS);
    float* pooled = alloc((size_t)N_GRAPHS * 512);

    const int B = 256;
    auto g = [](long n, int b) { return (unsigned)((n + b - 1) / b); };

    // init accumulators / maxima
    fill_f32<<<g(N_NODES * 8, B), B, 0, stream>>>(m1, N_NODES * 8, -INFINITY);
    fill_f32<<<g(N_NODES * 8, B), B, 0, stream>>>(denom1, N_NODES * 8, 0.f);
    fill_f32<<<g((long)N_NODES * 64, B), B, 0, stream>>>(out1, N_NODES * 64, 0.f);
    fill_f32<<<g(N_NODES, B), B, 0, stream>>>(m2, N_NODES, -INFINITY);
    fill_f32<<<g(N_NODES, B), B, 0, stream>>>(denom2, N_NODES, 0.f);
    fill_f32<<<g((long)N_NODES * 512, B), B, 0, stream>>>(out2, N_NODES * 512, 0.f);
    fill_f32<<<g(N_GRAPHS * 512, B), B, 0, stream>>>(sums, N_GRAPHS * 512, 0.f);
    fill_f32<<<g(N_GRAPHS, B), B, 0, stream>>>(cnt, N_GRAPHS, 0.f);

    // layer 1
    node1_kernel<<<g((long)N_NODES * 8, B), B, 0, stream>>>(x, W1, a_src1, a_dst1, h1, al_s1, al_d1);
    edge1_pass1 <<<g((long)N_EDGES * 8, B), B, 0, stream>>>(ei, al_s1, al_d1, e1, m1);
    edge1_pass2 <<<g((long)N_EDGES * 8, B), B, 0, stream>>>(ei, m1, e1, denom1);
    edge1_scatter<<<g((long)N_EDGES * 8, B), B, 0, stream>>>(ei, e1, denom1, h1, out1);
    act1_kernel <<<g((long)N_NODES * 64, B), B, 0, stream>>>(out1, b1);

    // layer 2: GEMM (WMMA f32) + attention dots
    gemm1_wmma<<<dim3(N_NODES / 16, 512 / (16 * 8)), B, 0, stream>>>(out1, W2, h2);
    node2_kernel<<<g(N_NODES, 8), B, 0, stream>>>(h2, a_src2, a_dst2, al_s2, al_d2);
    edge2_pass1<<<g(N_EDGES, B), B, 0, stream>>>(ei, al_s2, al_d2, e2, m2);
    edge2_pass2<<<g(N_EDGES, B), B, 0, stream>>>(ei, m2, e2, denom2);
    edge2_scatter<<<N_EDGES, 128, 0, stream>>>(ei, e2, denom2, h2, out2);

    // activation + pooling + head GEMM
    act2_pool<<<g((long)N_NODES * 512, B), B, 0, stream>>>(out2, b2, batch, sums);
    count_kernel<<<g(N_NODES, B), B, 0, stream>>>(batch, cnt);
    pooled_kernel<<<g(N_GRAPHS * 512, B), B, 0, stream>>>(sums, cnt, pooled);
    gemm2_wmma<<<16, B, 0, stream>>>(pooled, Wo, bo, out);
}